// GNN_SampleConcrete_24567212933209
// MI455X (gfx1250) — compile-verified
//
#include <hip/hip_runtime.h>
#include <math.h>

// Segmented (per-graph) Gumbel-softmax sampling for MI455X (gfx1250, wave32).
//
// x:        [N]    node logits (float32)
// uniforms: [N,30] uniforms in (0,1) (float32)
// ptr:      [B+1]  segment offsets (int32), ptr[0]=0, ptr[B]=N
// out:      [N]    max over 30 columns of segmented softmax probs
//
// One workgroup per graph; lane = column (30/32 active), wave = node.
// Pass 1: stage uniforms chunk global->LDS via GLOBAL_LOAD_ASYNC_TO_LDS_B32
//         (ASYNCcnt), compute online per-column (max, sum-exp), overwrite the
//         thread-private LDS slot with v = x - log(-log(u)).
// Pass 2: segments that fit the 480-node LDS cache read v back from LDS
//         (no global re-read, no transcendental recompute); longer segments
//         recompute from (L2-resident) global memory.

#define K3        30            // 3*K columns
#define BLOCK     256
#define WAVES     8             // BLOCK / 32
#define VCAP      480           // nodes cached in LDS: 480*32*4B = 60 KB

#if defined(__AMDGCN__) && __has_builtin(__builtin_amdgcn_global_load_async_to_lds_b32)
#define USE_ASYNC 1
#else
#define USE_ASYNC 0
#endif

// Builtin prototype (from hipcc diagnostic): first param is AS1 (global) int*,
// second is AS3 (LDS) int*, then imm offset, imm cpol.
typedef __attribute__((address_space(1))) int glb_i32;
typedef __attribute__((address_space(3))) int lds_i32;

__device__ __forceinline__ void wait_async0() {
#if __has_builtin(__builtin_amdgcn_s_wait_asynccnt)
    __builtin_amdgcn_s_wait_asynccnt(0);
#else
    asm volatile("s_wait_asynccnt 0x0" ::: "memory");
#endif
    asm volatile("" ::: "memory");   // compiler ordering fence
}

__global__ __launch_bounds__(BLOCK)
void seg_gumbel_softmax(const float* __restrict__ x,
                        const float* __restrict__ uni,
                        const int*   __restrict__ ptr,
                        float*       __restrict__ out)
{
    const int g     = blockIdx.x;
    const int start = ptr[g];
    const int end   = ptr[g + 1];
    if (end <= start) return;                 // empty segment

    const int  tid  = threadIdx.x;
    const int  wave = tid >> 5;
    const int  lane = tid & 31;
    const bool colv = (lane < K3);

    __shared__ float cache[VCAP * 32];        // 60 KB: u staged in, v written back
    __shared__ float sm_m[BLOCK];
    __shared__ float sm_s[BLOCK];
    __shared__ float sM[32];
    __shared__ float sSinv[32];

    const int  L    = end - start;
    const bool fits = (L <= VCAP);            // pass-2 can use the LDS v-cache

    // ---------------- Pass 1: online (max, sum-exp) per column ----------------
    float m = -INFINITY;
    float s = 0.0f;

    for (int cbase = start; cbase < end; cbase += VCAP) {
        const int cend = (cbase + VCAP < end) ? (cbase + VCAP) : end;

#if USE_ASYNC
        // Stage this chunk of uniforms into LDS with async DMA (per-lane,
        // EXEC-masked so columns 30/31 transfer nothing).
        if (colv) {
            for (int node = cbase + wave; node < cend; node += WAVES) {
                const float* gp = &uni[(size_t)node * K3 + lane];
                const float* lp = &cache[(size_t)(node - cbase) * 32 + lane];
                // flat->AS3: low 32 bits of a flat LDS address are the LDS
                // byte offset (aperture lives in the high bits).
                __builtin_amdgcn_global_load_async_to_lds_b32(
                    (glb_i32*)(uintptr_t)gp,
                    (lds_i32*)(uint32_t)(uintptr_t)lp,
                    0, 0);
            }
        }
        wait_async0();        // per-wave; each thread reads only its own slots
#endif

        for (int node = cbase + wave; node < cend; node += WAVES) {
            if (colv) {
                const int slot = (node - cbase) * 32 + lane;
#if USE_ASYNC
                const float u = cache[slot];
#else
                const float u = uni[(size_t)node * K3 + lane];
                if (node + WAVES < cend)     // gfx1250 global_prefetch_b8
                    __builtin_prefetch(&uni[(size_t)(node + WAVES) * K3 + lane], 0, 3);
#endif
                const float v = x[node] - logf(-logf(u));   // TAU == 1
                if (v > m) {
                    s = s * expf(m - v) + 1.0f;  // expf(-inf)=0 on first hit
                    m = v;
                } else {
                    s += expf(v - m);
                }
                cache[slot] = v;   // thread-private slot; valid for pass 2 iff fits
            }
        }
    }

    sm_m[tid] = m;
    sm_s[tid] = s;
    __syncthreads();

    // -------- Combine the 8 wave partials per column (first wave only) --------
    if (tid < 32) {
        float M = -INFINITY;
        #pragma unroll
        for (int w = 0; w < WAVES; ++w)
            M = fmaxf(M, sm_m[w * 32 + tid]);
        float S = 0.0f;
        #pragma unroll
        for (int w = 0; w < WAVES; ++w) {
            const float sw = sm_s[w * 32 + tid];
            if (sw > 0.0f)                      // skip -inf partials (avoid NaN)
                S += sw * expf(sm_m[w * 32 + tid] - M);
        }
        sM[tid]    = M;
        sSinv[tid] = (S > 0.0f) ? (1.0f / S) : 0.0f;
    }
    __syncthreads();

    const float Mc    = sM[lane];
    const float Sinvc = sSinv[lane];

    // --------- Pass 2: probs + per-node max over columns (1 node/wave) --------
    for (int node = start + wave; node < end; node += WAVES) {
        float p = 0.0f;                         // probs are strictly > 0
        if (colv) {
            float v;
            if (fits) {
                v = cache[(node - start) * 32 + lane];       // cached v
            } else {
                const float u = uni[(size_t)node * K3 + lane];  // L2-resident
                v = x[node] - logf(-logf(u));
            }
            p = expf(v - Mc) * Sinvc;           // v <= Mc always
        }
        // wave32 max-reduction across the 32 lanes (one node per wave)
        #pragma unroll
        for (int off = 16; off > 0; off >>= 1)
            p = fmaxf(p, __shfl_xor(p, off, 32));
        if (lane == 0)
            out[node] = p;
    }
}

extern "C" void kernel_launch(void* const* d_in, const int* in_sizes, int n_in,
                              void* d_out, int out_size, void* d_ws, size_t ws_size,
                              hipStream_t stream) {
    const float* x   = (const float*)d_in[0];   // [N]
    const float* uni = (const float*)d_in[1];   // [N, 30]
    const int*   ptr = (const int*)d_in[2];     // [B+1]
    float*       out = (float*)d_out;           // [N]

    const int B = in_sizes[2] - 1;              // number of graphs / segments
    seg_gumbel_softmax<<<dim3(B), dim3(BLOCK), 0, stream>>>(x, uni, ptr, out);
}